// QuantumDepthwiseQuanvolutionFilter_65481071401919
// MI455X (gfx1250) — compile-verified
//
#include <hip/hip_runtime.h>
#include <hip/hip_bf16.h>

// CDNA5 / gfx1250. Quanvolution collapsed to per-patch quadratic forms:
//   out[p,o] = s0(p)^T B_o s0(p) + bias_o + x[b,0,2i,2j]
// s0 = product state after data-dependent RY layer; B_o = sum_q pw[o,q] A_q
// precomputed from the fixed circuit unitary U. The P x 16 x 16 GEMM S*B_o
// runs on V_WMMA_F32_16X16X4_F32 (exact f32 semantics, K=16 -> 4 k-steps).
//
// Round-2: hot-kernel transcendentals -> v_sin/v_cos fast path (args in
// [0,0.5), deep inside fast domain); T-staging rows padded to 20 floats so
// the row-dot reads are 16B-aligned ds_load_b128 and still bank-conflict-free.

typedef float v2f __attribute__((ext_vector_type(2)));
typedef float v8f __attribute__((ext_vector_type(8)));

#define WAVES 8
#define TPAD  20   // row pad (floats): 80B row stride -> 16B aligned, 20r%64 distinct

// ---------------------------------------------------------------------------
// Kernel 1: build B_o[4][16][16] in workspace from q_weights + pw_weight.
// Lanes 0..15 each simulate one basis column of the fixed unitary
// U = (CNOTchain * RXlayer)^2 ; then 256 threads form
// A_q[j,k] = sum_i z_q(i) (ReU_ij ReU_ik + ImU_ij ImU_ik),  B_o = sum_q pw A_q.
// Runs once; full-precision sinf/cosf kept here since B feeds every patch.
// ---------------------------------------------------------------------------
__global__ __launch_bounds__(256) void quanv_setup_kernel(
    const float* __restrict__ qw,    // (2,4,1)
    const float* __restrict__ pww,   // (4,4,1,1)
    float* __restrict__ Bws)         // out: 4*16*16 f32
{
  __shared__ float Ure[16][16];  // [col j][row i]
  __shared__ float Uim[16][16];
  const int t = threadIdx.x;

  if (t < 16) {
    float re[16], im[16];
#pragma unroll
    for (int i = 0; i < 16; ++i) { re[i] = 0.f; im[i] = 0.f; }
    re[t] = 1.f;  // basis state |j=t>

#pragma unroll
    for (int l = 0; l < 2; ++l) {
      // RX layer: gate [[c, -i s],[-i s, c]] on qubit q (bit position 3-q)
#pragma unroll
      for (int q = 0; q < 4; ++q) {
        const float th = 0.5f * qw[l * 4 + q];
        const float c = cosf(th), s = sinf(th);
        const int bit = 1 << (3 - q);
#pragma unroll
        for (int i0 = 0; i0 < 16; ++i0) {
          if (!(i0 & bit)) {
            const int i1 = i0 | bit;
            const float r0 = re[i0], m0 = im[i0];
            const float r1 = re[i1], m1 = im[i1];
            re[i0] = c * r0 + s * m1;  im[i0] = c * m0 - s * r1;
            re[i1] = s * m0 + c * r1;  im[i1] = c * m1 - s * r0;
          }
        }
      }
      // CNOT chain: control q -> target q+1 (swap target bit where control=1)
#pragma unroll
      for (int q = 0; q < 3; ++q) {
        const int cb = 1 << (3 - q), tb = 1 << (2 - q);
#pragma unroll
        for (int i0 = 0; i0 < 16; ++i0) {
          if ((i0 & cb) && !(i0 & tb)) {
            const int i1 = i0 | tb;
            float tr = re[i0]; re[i0] = re[i1]; re[i1] = tr;
            float ti = im[i0]; im[i0] = im[i1]; im[i1] = ti;
          }
        }
      }
    }
#pragma unroll
    for (int i = 0; i < 16; ++i) { Ure[t][i] = re[i]; Uim[t][i] = im[i]; }
  }
  __syncthreads();

  // 256 threads == 256 (j,k) entries
  const int j = t >> 4, k = t & 15;
  float aq[4] = {0.f, 0.f, 0.f, 0.f};
#pragma unroll
  for (int i = 0; i < 16; ++i) {
    const float d = Ure[j][i] * Ure[k][i] + Uim[j][i] * Uim[k][i];
    aq[0] += (i & 8) ? -d : d;   // qubit 0 = MSB
    aq[1] += (i & 4) ? -d : d;
    aq[2] += (i & 2) ? -d : d;
    aq[3] += (i & 1) ? -d : d;
  }
#pragma unroll
  for (int o = 0; o < 4; ++o) {
    float v = 0.f;
#pragma unroll
    for (int q = 0; q < 4; ++q) v += pww[o * 4 + q] * aq[q];
    Bws[(o * 16 + j) * 16 + k] = v;
  }
}

// ---------------------------------------------------------------------------
// Kernel 2: main. One 16-patch tile per wave; 16x v_wmma_f32_16x16x4_f32
// per tile (4 channels x 4 K-chunks), then row-dot via LDS-staged T.
// ---------------------------------------------------------------------------
__global__ __launch_bounds__(256) void quanv_main_kernel(
    const float* __restrict__ x,     // (B,1,28,28)
    const float* __restrict__ Bws,   // 4*16*16
    const float* __restrict__ bias,  // (4,)
    float* __restrict__ out,         // (B,4,14,14)
    int Bimgs)
{
  __shared__ float Bls[4 * 16 * 16];
  __shared__ __align__(16) float Tls[WAVES][4][16][TPAD];

  const int tid = threadIdx.x;
  for (int idx = tid; idx < 4 * 16 * 16; idx += 256) Bls[idx] = Bws[idx];
  __syncthreads();

  const int lane = tid & 31;
  const int wave = tid >> 5;
  const int hi   = lane >> 4;   // half-wave select (K pair / row half / chan pair)
  const int r    = lane & 15;   // patch-in-tile == M row == N col

  const unsigned P    = (unsigned)Bimgs * 196u;
  const unsigned tile = (unsigned)blockIdx.x * WAVES + (unsigned)wave;
  const unsigned p    = tile * 16u + (unsigned)r;
  const bool valid    = (p < P);
  const unsigned pc   = valid ? p : (P - 1u);

  const unsigned b   = pc / 196u;        // const-divisor -> mulhi+shift
  const unsigned rem = pc % 196u;
  const unsigned pi  = rem / 14u, pj = rem % 14u;

  const float* xp = x + (size_t)b * 784u + (2u * pi) * 28u + 2u * pj;
  __builtin_prefetch(xp, 0, 1);                 // global_prefetch_b8
  const float p0 = xp[0],  p1 = xp[1];
  const float p2 = xp[28], p3 = xp[29];

  // s0_j = prod_q (bit_q(j) ? sin : cos)(p_q/2), qubit0 = MSB of j.
  // Args are x/2 in [0,0.5): hardware v_sin/v_cos fast path is exact enough
  // (no range reduction / special-case code needed).
  const float c0 = __cosf(0.5f * p0), s0_ = __sinf(0.5f * p0);
  const float c1 = __cosf(0.5f * p1), s1_ = __sinf(0.5f * p1);
  const float c2 = __cosf(0.5f * p2), s2_ = __sinf(0.5f * p2);
  const float c3 = __cosf(0.5f * p3), s3_ = __sinf(0.5f * p3);
  const float m01[4] = {c0 * c1, c0 * s1_, s0_ * c1, s0_ * s1_};
  const float m23[4] = {c2 * c3, c2 * s3_, s2_ * c3, s2_ * s3_};

  // Full s0 (constant indices only -> stays in VGPRs) for the final row-dot.
  float sv[16];
#pragma unroll
  for (int n = 0; n < 16; ++n) sv[n] = m01[n >> 2] * m23[n & 3];

  // A fragments (16x4 f32, ISA 7.12.2): lanes 0-15 hold K={0,1}, 16-31 K={2,3}.
  // For j = 4*ki + 2*hi + e:  j>>2 == ki (constant), j&3 == 2*hi + e, so the
  // runtime part collapses to two 2-way selects instead of sv[runtime] chains.
  const float m23lo = hi ? m23[2] : m23[0];
  const float m23hi = hi ? m23[3] : m23[1];
  v2f aF[4];
#pragma unroll
  for (int ki = 0; ki < 4; ++ki) {
    aF[ki].x = m01[ki] * m23lo;
    aF[ki].y = m01[ki] * m23hi;
  }

  // GEMM: T_o = S * B_o  (K = 16 -> 4 k-steps of 16x16x4)
  v8f acc[4];
#pragma unroll
  for (int o = 0; o < 4; ++o) {
    v8f c = {0.f, 0.f, 0.f, 0.f, 0.f, 0.f, 0.f, 0.f};
#pragma unroll
    for (int ki = 0; ki < 4; ++ki) {
      const int row = 4 * ki + 2 * hi;  // B frag mirrors A layout (assumed)
      v2f bF;
      bF.x = Bls[(o * 16 + row) * 16 + r];      // LDS dynamic addr: fine
      bF.y = Bls[(o * 16 + row + 1) * 16 + r];
      c = __builtin_amdgcn_wmma_f32_16x16x4_f32(
          /*neg_a=*/false, aF[ki], /*neg_b=*/false, bF,
          /*c_mod=*/(short)0, c, /*reuse_a=*/false, /*reuse_b=*/false);
    }
    acc[o] = c;
  }

  // C layout: VGPR v -> row (v + 8*hi), col r. Stage to LDS for the row-dot.
#pragma unroll
  for (int o = 0; o < 4; ++o)
#pragma unroll
    for (int v = 0; v < 8; ++v)
      Tls[wave][o][v + 8 * hi][r] = acc[o][v];
  __syncthreads();

  // y_o(p) = sum_n s0_n * T_o[p, n] + bias_o + residual (x[b,0,2i,2j] == p0)
  const float bb0 = bias[0], bb1 = bias[1], bb2 = bias[2], bb3 = bias[3];
  const float biasA = hi ? bb2 : bb0;   // channel 2*hi
  const float biasB = hi ? bb3 : bb1;   // channel 2*hi + 1
  const int   oA = 2 * hi, oB = 2 * hi + 1;

  // Rows are 16B-aligned (TPAD=20 -> 80B stride): b128 LDS reads.
  const float4* rowA = (const float4*)&Tls[wave][oA][r][0];
  const float4* rowB = (const float4*)&Tls[wave][oB][r][0];
  float yA = 0.f, yB = 0.f;
#pragma unroll
  for (int q4 = 0; q4 < 4; ++q4) {
    const float4 ta = rowA[q4];
    const float4 tb = rowB[q4];
    yA += sv[4 * q4 + 0] * ta.x + sv[4 * q4 + 1] * ta.y +
          sv[4 * q4 + 2] * ta.z + sv[4 * q4 + 3] * ta.w;
    yB += sv[4 * q4 + 0] * tb.x + sv[4 * q4 + 1] * tb.y +
          sv[4 * q4 + 2] * tb.z + sv[4 * q4 + 3] * tb.w;
  }
  yA += biasA + p0;
  yB += biasB + p0;

  if (valid) {
    const unsigned base = b * 784u + rem;       // (b*4 + o)*196 + rem
    out[base + (unsigned)oA * 196u] = yA;
    out[base + (unsigned)oB * 196u] = yB;
  }
}

// ---------------------------------------------------------------------------
extern "C" void kernel_launch(void* const* d_in, const int* in_sizes, int n_in,
                              void* d_out, int out_size, void* d_ws, size_t ws_size,
                              hipStream_t stream) {
  const float* x   = (const float*)d_in[0];  // (B,1,28,28)
  const float* qw  = (const float*)d_in[1];  // (2,4,1)
  const float* pww = (const float*)d_in[2];  // (4,4,1,1)
  const float* pwb = (const float*)d_in[3];  // (4,)
  float* out = (float*)d_out;
  float* Bws = (float*)d_ws;                 // 4*16*16 f32 = 4 KB

  const int Bimgs = in_sizes[0] / 784;
  const long P = (long)Bimgs * 196;
  const long tiles = (P + 15) / 16;
  const int blocks = (int)((tiles + WAVES - 1) / WAVES);

  quanv_setup_kernel<<<1, 256, 0, stream>>>(qw, pww, Bws);
  quanv_main_kernel<<<blocks, 256, 0, stream>>>(x, Bws, pwb, out, Bimgs);
}